// SDEGenerator_35373350650571
// MI455X (gfx1250) — compile-verified
//
#include <hip/hip_runtime.h>
#include <stdint.h>

// ---------------------------------------------------------------------------
// SDE Euler-Maruyama generator for MI455X (gfx1250)
//   x_{k+1} = x_k + (A x_k + bA) dt + (C x_k + bC) * (sqrt(dt) * eps_k)
// B=4096 paths, T=2500 steps, D=3.  Memory-bound (~246 MB @ 23.3 TB/s).
// Noise is streamed into LDS by the CDNA5 Tensor Data Mover (TDM):
// each wave issues one tensor_load_to_lds per chunk for its 2D subtile
// (5 rows x 384 floats, row stride B*3), tracked with TENSORcnt and
// double-buffered against compute.  Output is staged in LDS to coalesce
// the [B, T+1, 3] strided layout.
// ---------------------------------------------------------------------------

#define BLK    128               // threads per block (4 wave32)
#define WAVES  (BLK / 32)        // 4 waves, one TDM subtile each
#define TC     20                // time steps per chunk
#define TCW    (TC / WAVES)      // 5 rows per wave's TDM subtile
#define TC3    (TC * 3)          // 60 output dwords per path per chunk
#define PAD    (TC3 + 1)         // 61: padded LDS row (gcd(61,64)=1 -> no bank conflicts)
#define NB     4096              // batch (paths)
#define NT     2500              // time steps
#define NCH    (NT / TC)         // 125 chunks
#define BD3    (NB * 3)          // dwords per timestep slab of noise (12288)
#define TILE_W (BLK * 3)         // 384 floats per noise row per block
#define R3     ((NT + 1) * 3)    // dwords per output row (7503)
#define DT_F   0.002f
#define SQDT_F 0.044721359549995794f   // sqrt(0.002)

typedef unsigned int u32;
typedef u32 v4u __attribute__((ext_vector_type(4)));
typedef u32 v8u __attribute__((ext_vector_type(8)));

// TDM 2D tile load: copies rows x TILE_W floats from gaddr to LDS at byte address lds_off,
// row stride BD3 floats.  D# per CDNA5 ISA §8.3/§8.4; 2D tensor -> 2 SGPR
// groups (VADDR2/VADDR3 omitted).  Tracked by TENSORcnt.
__device__ __forceinline__ void tdm_load_2d(u32 lds_off, const float* gaddr, u32 rows) {
    const unsigned long long ga = (unsigned long long)(uintptr_t)gaddr;

    v4u g0;
    g0.x = 1u;                                            // count=1, user D#
    g0.y = lds_off;                                       // lds_addr (bytes)
    g0.z = (u32)ga;                                       // global_addr[31:0]
    g0.w = (u32)((ga >> 32) & 0x01FFFFFFu) | (2u << 30);  // addr[56:32] | type=2

    v8u g1;
    g1.s0 = 2u << 16;                                     // data_size = 4 bytes
    g1.s1 = ((u32)TILE_W & 0xFFFFu) << 16;                // tensor_dim0[15:0] << 16
    g1.s2 = ((rows & 0xFFFFu) << 16) | (((u32)TILE_W >> 16) & 0xFFFFu); // td1 lo | td0 hi
    g1.s3 = ((u32)TILE_W & 0xFFFFu) << 16;                // tile_dim0 << 16 | td1 hi(0)
    g1.s4 = rows & 0xFFFFu;                               // tile_dim1 (tile_dim2=0)
    g1.s5 = (u32)BD3;                                     // tensor_dim0_stride[31:0]
    g1.s6 = 0u;                                           // stride hi / dim1_stride lo
    g1.s7 = 0u;                                           // dim1_stride hi (unused, 2D)

    asm volatile("tensor_load_to_lds %0, %1"
                 :: "s"(g0), "s"(g1)
                 : "memory");
}

__device__ __forceinline__ void wait_tensor0() {
#if defined(__has_builtin)
#if __has_builtin(__builtin_amdgcn_s_wait_tensorcnt)
    __builtin_amdgcn_s_wait_tensorcnt(0);
#else
    asm volatile("s_wait_tensorcnt 0x0" ::: "memory");
#endif
#else
    asm volatile("s_wait_tensorcnt 0x0" ::: "memory");
#endif
}

__global__ __launch_bounds__(BLK)
void sde_euler_kernel(const float* __restrict__ x0,
                      const float* __restrict__ A,    // drift     [3][3]
                      const float* __restrict__ bA,   // drift_bias [3]
                      const float* __restrict__ C,    // diffusion [3][3]
                      const float* __restrict__ bC,   // diffusion_bias [3]
                      const float* __restrict__ noise,// [T][B][3]
                      float* __restrict__ out)        // [B][T+1][3]
{
    __shared__ float nbuf[2][TC * TILE_W]; // 2 x 30 KB noise double buffer
    __shared__ float obuf[BLK * PAD];      // 30.5 KB output staging

    const int tid = threadIdx.x;
    const int wid = tid >> 5;              // wave id (uniform per wave)
    const int b0  = blockIdx.x * BLK;
    const int b   = b0 + tid;

    // Per-wave TDM subtile offsets (wave-uniform).
    const size_t g_sub = (size_t)wid * TCW * BD3;          // global dword offset
    const u32    l_sub = (u32)(wid * TCW * TILE_W * 4);    // LDS byte offset

    // Uniform model parameters -> scalar loads / broadcast.
    const float a00 = A[0], a01 = A[1], a02 = A[2];
    const float a10 = A[3], a11 = A[4], a12 = A[5];
    const float a20 = A[6], a21 = A[7], a22 = A[8];
    const float ba0 = bA[0], ba1 = bA[1], ba2 = bA[2];
    const float c00 = C[0], c01 = C[1], c02 = C[2];
    const float c10 = C[3], c11 = C[4], c12 = C[5];
    const float c20 = C[6], c21 = C[7], c22 = C[8];
    const float bc0 = bC[0], bc1 = bC[1], bc2 = bC[2];

    float xv0 = x0[b * 3 + 0];
    float xv1 = x0[b * 3 + 1];
    float xv2 = x0[b * 3 + 2];

    // paths[b][0][:] = x0 (one-time scattered store, negligible)
    const size_t ob = (size_t)b * R3;
    out[ob + 0] = xv0;
    out[ob + 1] = xv1;
    out[ob + 2] = xv2;

    // Prologue: every wave DMAs its subtile of chunk 0.
    tdm_load_2d((u32)(uintptr_t)&nbuf[0][0] + l_sub,
                noise + (size_t)b0 * 3 + g_sub, TCW);

    for (int c = 0; c < NCH; ++c) {
        wait_tensor0();       // this wave's subtile of chunk c landed in LDS
        __syncthreads();      // ...and the other waves'; buffer reuse safe

        if (c + 1 < NCH) {    // DMA next chunk while we compute this one
            tdm_load_2d((u32)(uintptr_t)&nbuf[(c + 1) & 1][0] + l_sub,
                        noise + (size_t)(c + 1) * TC * BD3 + (size_t)b0 * 3 + g_sub,
                        TCW);
        }

        const float* nb = &nbuf[c & 1][0];
#pragma unroll
        for (int t = 0; t < TC; ++t) {
            const float e0 = nb[t * TILE_W + tid * 3 + 0];
            const float e1 = nb[t * TILE_W + tid * 3 + 1];
            const float e2 = nb[t * TILE_W + tid * 3 + 2];
            const float dw0 = SQDT_F * e0;
            const float dw1 = SQDT_F * e1;
            const float dw2 = SQDT_F * e2;

            const float y10 = fmaf(a00, xv0, fmaf(a01, xv1, fmaf(a02, xv2, ba0)));
            const float y11 = fmaf(a10, xv0, fmaf(a11, xv1, fmaf(a12, xv2, ba1)));
            const float y12 = fmaf(a20, xv0, fmaf(a21, xv1, fmaf(a22, xv2, ba2)));
            const float y20 = fmaf(c00, xv0, fmaf(c01, xv1, fmaf(c02, xv2, bc0)));
            const float y21 = fmaf(c10, xv0, fmaf(c11, xv1, fmaf(c12, xv2, bc1)));
            const float y22 = fmaf(c20, xv0, fmaf(c21, xv1, fmaf(c22, xv2, bc2)));

            xv0 = fmaf(y10, DT_F, fmaf(y20, dw0, xv0));
            xv1 = fmaf(y11, DT_F, fmaf(y21, dw1, xv1));
            xv2 = fmaf(y12, DT_F, fmaf(y22, dw2, xv2));

            obuf[tid * PAD + t * 3 + 0] = xv0;   // stride-61: conflict-free
            obuf[tid * PAD + t * 3 + 1] = xv1;
            obuf[tid * PAD + t * 3 + 2] = xv2;
        }

        __syncthreads();      // staged results visible to all waves

        // Coalesced writeback: out[b0+row][c*TC+1 .. c*TC+TC][:] per row,
        // consecutive lanes -> consecutive dwords.
        const int gt0 = (c * TC + 1) * 3;
#pragma unroll 4
        for (int k = 0; k < TC3; ++k) {
            const int g   = k * BLK + tid;
            const int row = g / TC3;
            const int col = g - row * TC3;
            out[(size_t)(b0 + row) * R3 + gt0 + col] = obuf[row * PAD + col];
        }
    }
}

extern "C" void kernel_launch(void* const* d_in, const int* in_sizes, int n_in,
                              void* d_out, int out_size, void* d_ws, size_t ws_size,
                              hipStream_t stream) {
    (void)in_sizes; (void)n_in; (void)out_size; (void)d_ws; (void)ws_size;
    const float* x0     = (const float*)d_in[0];
    const float* drift  = (const float*)d_in[1];
    const float* driftb = (const float*)d_in[2];
    const float* diff   = (const float*)d_in[3];
    const float* diffb  = (const float*)d_in[4];
    const float* noise  = (const float*)d_in[5];
    float* out = (float*)d_out;

    dim3 grid(NB / BLK);   // 32 workgroups, one WGP each
    dim3 block(BLK);       // 4 wave32 per workgroup
    hipLaunchKernelGGL(sde_euler_kernel, grid, block, 0, stream,
                       x0, drift, driftb, diff, diffb, noise, out);
}